// GATModel_652835029252
// MI455X (gfx1250) — compile-verified
//
#include <hip/hip_runtime.h>
#include <hip/hip_bf16.h>
#include <math.h>

namespace {

constexpr int NN   = 50000;   // nodes
constexpr int BG   = 64;      // graphs
constexpr int EE   = 800000;  // edges (without self loops)
constexpr int ET   = EE + NN; // edges + self loops
constexpr int KDIM = 128;     // NODE_DIM == HID*HEADS
constexpr float BN_INV = 0.9999950000374997f; // 1/sqrt(1+1e-5)

typedef __attribute__((ext_vector_type(2))) float v2f;
typedef __attribute__((ext_vector_type(8))) float v8f;

// ---------- helpers ----------
__device__ __forceinline__ unsigned ordF(float f) {
  unsigned u = __float_as_uint(f);
  return (u & 0x80000000u) ? ~u : (u | 0x80000000u);  // order-preserving float->uint
}
__device__ __forceinline__ float unordF(unsigned u) {
  return __uint_as_float((u & 0x80000000u) ? (u & 0x7FFFFFFFu) : ~u);
}
__device__ __forceinline__ float eluf(float x)   { return x > 0.f ? x : expm1f(x); }
__device__ __forceinline__ float lrelu(float x)  { return x > 0.f ? x : 0.2f * x; }

__global__ void k_fill_u32(unsigned* __restrict__ p, unsigned v, int n) {
  int i = blockIdx.x * blockDim.x + threadIdx.x;
  if (i < n) p[i] = v;
}

// ---------- GEMM: C[NN x Nc] = A[NN x 128] * W[128 x Nc], f32 WMMA ----------
__global__ void k_gemm_wmma_f32(const float* __restrict__ A,
                                const float* __restrict__ W,
                                float* __restrict__ C, int Nc) {
  int wid  = (blockIdx.x * blockDim.x + threadIdx.x) >> 5;  // wave-uniform
  int lane = threadIdx.x & 31;
  int tilesN = Nc >> 4;
  int totalTiles = (NN >> 4) * tilesN;
  if (wid >= totalTiles) return;                             // whole wave exits
  int tm = (wid / tilesN) << 4;
  int tn = (wid % tilesN) << 4;
  int r  = lane & 15;            // A row / B col / C col within tile
  int kg = (lane >> 4) << 1;     // K sub-pair selector: 0 or 2
  v8f acc = {};
  const float* Arow = A + (size_t)(tm + r) * KDIM;
#pragma unroll 4
  for (int k0 = 0; k0 < KDIM; k0 += 4) {
    int ka = k0 + kg;
    v2f a, b;
    a.x = Arow[ka];
    a.y = Arow[ka + 1];
    b.x = W[(size_t)ka * Nc + tn + r];
    b.y = W[(size_t)(ka + 1) * Nc + tn + r];
#if __has_builtin(__builtin_amdgcn_wmma_f32_16x16x4_f32)
    acc = __builtin_amdgcn_wmma_f32_16x16x4_f32(false, a, false, b,
                                                (short)0, acc, false, false);
#else
    // scalar fallback (host pass / missing builtin): same math, direct loads
    int rb = (lane >> 4) << 3;
#pragma unroll
    for (int v = 0; v < 8; v++) {
      float s = acc[v];
#pragma unroll
      for (int kk = 0; kk < 4; kk++)
        s += A[(size_t)(tm + rb + v) * KDIM + k0 + kk] *
             W[(size_t)(k0 + kk) * Nc + tn + r];
      acc[v] = s;
    }
#endif
  }
  int rbase = (lane >> 4) << 3;  // C: vgpr v -> row v (lanes 0-15) / v+8 (lanes 16-31)
#pragma unroll
  for (int v = 0; v < 8; v++)
    C[(size_t)(tm + rbase + v) * Nc + tn + r] = acc[v];
}

// ---------- attention scalars: es/ed per (node, head) ----------
__global__ void k_esed(const float* __restrict__ hW, const float* __restrict__ as,
                       const float* __restrict__ ad, float* __restrict__ es,
                       float* __restrict__ ed, int H, int O) {
  int i = blockIdx.x * blockDim.x + threadIdx.x;
  if (i >= NN * H) return;
  int n = i / H, h = i - n * H;
  const float* hp = hW + (size_t)n * H * O + h * O;
  float s = 0.f, d = 0.f;
  for (int o = 0; o < O; o++) { float v = hp[o]; s += v * as[h * O + o]; d += v * ad[h * O + o]; }
  es[i] = s; ed[i] = d;
}

// ---------- per-edge segment max (ordered-uint atomicMax) ----------
__global__ void k_attn_max(const int* __restrict__ ei, const float* __restrict__ es,
                           const float* __restrict__ ed, unsigned* __restrict__ mI, int H) {
  int e = blockIdx.x * blockDim.x + threadIdx.x;
  if (e >= ET) return;
  int s = (e < EE) ? ei[e] : e - EE;
  int d = (e < EE) ? ei[EE + e] : e - EE;
  for (int h = 0; h < H; h++) {
    float ev = lrelu(es[s * H + h] + ed[d * H + h]);
    atomicMax(&mI[d * H + h], ordF(ev));
  }
}

// ---------- exp(e - m), accumulate denominator ----------
__global__ void k_attn_exp(const int* __restrict__ ei, const float* __restrict__ es,
                           const float* __restrict__ ed, const unsigned* __restrict__ mI,
                           float* __restrict__ den, float* __restrict__ wB, int H) {
  int e = blockIdx.x * blockDim.x + threadIdx.x;
  if (e >= ET) return;
  int s = (e < EE) ? ei[e] : e - EE;
  int d = (e < EE) ? ei[EE + e] : e - EE;
  for (int h = 0; h < H; h++) {
    float ev = lrelu(es[s * H + h] + ed[d * H + h]);
    float w = expf(ev - unordF(mI[d * H + h]));
    wB[e * H + h] = w;
    atomicAdd(&den[d * H + h], w);
  }
}

// ---------- normalize weights in place: w -> alpha ----------
__global__ void k_attn_alpha(const int* __restrict__ ei, const float* __restrict__ den,
                             float* __restrict__ wB, int H) {
  int i = blockIdx.x * blockDim.x + threadIdx.x;
  if (i >= ET * H) return;
  int e = i / H, h = i - e * H;
  int d = (e < EE) ? ei[EE + e] : e - EE;
  wB[i] = wB[i] / (den[d * H + h] + 1e-16f);
}

// ---------- weighted scatter-add aggregation ----------
__global__ void k_attn_agg(const int* __restrict__ ei, const float* __restrict__ hW,
                           const float* __restrict__ alpha, float* __restrict__ agg,
                           int H, int O, int HO) {
  int i = blockIdx.x * blockDim.x + threadIdx.x;
  if (i >= ET * HO) return;
  int e = i / HO, dd = i - e * HO, h = dd / O;
  int s = (e < EE) ? ei[e] : e - EE;
  int d = (e < EE) ? ei[EE + e] : e - EE;
  atomicAdd(&agg[(size_t)d * HO + dd], hW[(size_t)s * HO + dd] * alpha[e * H + h]);
}

// ---------- bias + (residual) + BN-eval + ELU ----------
__global__ void k_bnelu(const float* __restrict__ agg, const float* __restrict__ bias,
                        const float* __restrict__ g, const float* __restrict__ be,
                        const float* __restrict__ res, int hasRes,
                        float* __restrict__ out, int C) {
  int i = blockIdx.x * blockDim.x + threadIdx.x;
  if (i >= NN * C) return;
  int c = i % C;
  float x = agg[i] + bias[c];
  if (hasRes) x += res[i];
  x = g[c] * x * BN_INV + be[c];
  out[i] = eluf(x);
}

// ---------- mean pool (sum + count) ----------
__global__ void k_pool(const int* __restrict__ batch, const float* __restrict__ h2,
                       float* __restrict__ sums, float* __restrict__ cnt) {
  int i = blockIdx.x * blockDim.x + threadIdx.x;
  if (i >= NN * 32) return;
  int n = i >> 5, d = i & 31;
  int g = batch[n];
  atomicAdd(&sums[g * 32 + d], h2[i]);
  if (d == 0) atomicAdd(&cnt[g], 1.f);
}

// ---------- tiny conv/MLP head, one block per graph ----------
__global__ void k_head(const float* __restrict__ sums, const float* __restrict__ cnt,
                       const float* __restrict__ b2,
                       const float* __restrict__ c0w, const float* __restrict__ c0b,
                       const float* __restrict__ cg0, const float* __restrict__ cb0,
                       const float* __restrict__ c1w, const float* __restrict__ c1b,
                       const float* __restrict__ l1w, const float* __restrict__ l1b,
                       const float* __restrict__ l2w, const float* __restrict__ l2b,
                       float* __restrict__ outLog, float* __restrict__ outEmb) {
  int b = blockIdx.x, tid = threadIdx.x;
  __shared__ float p[32];
  __shared__ float z1[32][33];
  __shared__ float emb[16];
  __shared__ float y1[8];
  __shared__ float lg[10];
  if (tid < 32) {
    float c = cnt[b]; c = c < 1.f ? 1.f : c;
    p[tid] = sums[b * 32 + tid] / c + b2[tid];
  }
  __syncthreads();
  for (int i = tid; i < 1024; i += blockDim.x) {  // conv1 (C=1 in, 32 out) + BN + ELU
    int co = i >> 5, t = i & 31;
    float s = c0b[co];
#pragma unroll
    for (int k = 0; k < 5; k++) {
      int tt = t + k - 2;
      float v = (tt >= 0 && tt < 32) ? p[tt] : 0.f;
      s += v * c0w[co * 5 + k];
    }
    s = cg0[co] * s * BN_INV + cb0[co];
    z1[co][t] = eluf(s);
  }
  __syncthreads();
  if (tid < 16) {                                   // conv2 + mean over t
    int co = tid;
    float acc = 0.f;
    for (int t = 0; t < 32; t++) {
      float s = c1b[co];
      for (int ci = 0; ci < 32; ci++) {
#pragma unroll
        for (int k = 0; k < 5; k++) {
          int tt = t + k - 2;
          float v = (tt >= 0 && tt < 32) ? z1[ci][tt] : 0.f;
          s += v * c1w[co * 160 + ci * 5 + k];
        }
      }
      acc += s;
    }
    float e = acc * (1.f / 32.f);
    emb[co] = e;
    outEmb[b * 16 + co] = e;
  }
  __syncthreads();
  if (tid < 8) {                                    // linear1 + ELU
    float s = l1b[tid];
    for (int i = 0; i < 16; i++) s += emb[i] * l1w[i * 8 + tid];
    y1[tid] = eluf(s);
  }
  __syncthreads();
  if (tid < 10) {                                   // linear2
    float s = l2b[tid];
    for (int j = 0; j < 8; j++) s += y1[j] * l2w[j * 10 + tid];
    lg[tid] = s;
  }
  __syncthreads();
  if (tid == 0) {                                   // log_softmax
    float mx = lg[0];
    for (int c = 1; c < 10; c++) mx = fmaxf(mx, lg[c]);
    float se = 0.f;
    for (int c = 0; c < 10; c++) se += expf(lg[c] - mx);
    float lse = logf(se) + mx;
    for (int c = 0; c < 10; c++) outLog[b * 10 + c] = lg[c] - lse;
  }
}

inline int gridFor(long long n, int bs) { return (int)((n + bs - 1) / bs); }

} // namespace

extern "C" void kernel_launch(void* const* d_in, const int* in_sizes, int n_in,
                              void* d_out, int out_size, void* d_ws, size_t ws_size,
                              hipStream_t stream) {
  const float* x    = (const float*)d_in[0];
  const int*   ei   = (const int*)  d_in[1];
  const int*   batch= (const int*)  d_in[2];
  const float* W0   = (const float*)d_in[3];
  const float* a0s  = (const float*)d_in[4];
  const float* a0d  = (const float*)d_in[5];
  const float* b0   = (const float*)d_in[6];
  const float* W1   = (const float*)d_in[7];
  const float* a1s  = (const float*)d_in[8];
  const float* a1d  = (const float*)d_in[9];
  const float* b1   = (const float*)d_in[10];
  const float* W2   = (const float*)d_in[11];
  const float* a2s  = (const float*)d_in[12];
  const float* a2d  = (const float*)d_in[13];
  const float* b2   = (const float*)d_in[14];
  const float* g0   = (const float*)d_in[15];
  const float* be0  = (const float*)d_in[16];
  const float* g1   = (const float*)d_in[17];
  const float* be1  = (const float*)d_in[18];
  const float* c0w  = (const float*)d_in[19];
  const float* c0b  = (const float*)d_in[20];
  const float* cg0  = (const float*)d_in[21];
  const float* cb0  = (const float*)d_in[22];
  const float* c1w  = (const float*)d_in[23];
  const float* c1b  = (const float*)d_in[24];
  const float* l1w  = (const float*)d_in[25];
  const float* l1b  = (const float*)d_in[26];
  const float* l2w  = (const float*)d_in[27];
  const float* l2b  = (const float*)d_in[28];

  float* ws = (float*)d_ws;
  size_t off = 0;
  float*    hW   = ws + off; off += (size_t)NN * 128;
  float*    agg  = ws + off; off += (size_t)NN * 128;
  float*    h    = ws + off; off += (size_t)NN * 128;
  float*    es   = ws + off; off += (size_t)NN * 4;
  float*    ed   = ws + off; off += (size_t)NN * 4;
  unsigned* mI   = (unsigned*)(ws + off); off += (size_t)NN * 4;
  float*    den  = ws + off; off += (size_t)NN * 4;
  float*    wB   = ws + off; off += (size_t)ET * 4;
  float*    sums = ws + off; off += (size_t)BG * 32;
  float*    cnt  = ws + off; off += (size_t)BG;

  const int BS = 256;
  const unsigned ORD_NEG_INF = 0x007FFFFFu; // ordF(-inf)

  auto run_attention = [&](const float* as, const float* ad, int H, int O) {
    int HO = H * O;
    k_esed<<<gridFor((long long)NN * H, BS), BS, 0, stream>>>(hW, as, ad, es, ed, H, O);
    k_fill_u32<<<gridFor((long long)NN * H, BS), BS, 0, stream>>>(mI, ORD_NEG_INF, NN * H);
    k_fill_u32<<<gridFor((long long)NN * H, BS), BS, 0, stream>>>((unsigned*)den, 0u, NN * H);
    k_fill_u32<<<gridFor((long long)NN * HO, BS), BS, 0, stream>>>((unsigned*)agg, 0u, NN * HO);
    k_attn_max  <<<gridFor(ET, BS), BS, 0, stream>>>(ei, es, ed, mI, H);
    k_attn_exp  <<<gridFor(ET, BS), BS, 0, stream>>>(ei, es, ed, mI, den, wB, H);
    k_attn_alpha<<<gridFor((long long)ET * H, BS), BS, 0, stream>>>(ei, den, wB, H);
    k_attn_agg  <<<gridFor((long long)ET * HO, BS), BS, 0, stream>>>(ei, hW, wB, agg, H, O, HO);
  };

  // ---- GAT layer 0 ----
  {
    int tiles = (NN / 16) * (128 / 16);
    k_gemm_wmma_f32<<<gridFor((long long)tiles * 32, BS), BS, 0, stream>>>(x, W0, hW, 128);
    run_attention(a0s, a0d, 4, 32);
    k_bnelu<<<gridFor((long long)NN * 128, BS), BS, 0, stream>>>(agg, b0, g0, be0, nullptr, 0, h, 128);
  }
  // ---- GAT layer 1 (residual) ----
  {
    int tiles = (NN / 16) * (128 / 16);
    k_gemm_wmma_f32<<<gridFor((long long)tiles * 32, BS), BS, 0, stream>>>(h, W1, hW, 128);
    run_attention(a1s, a1d, 4, 32);
    k_bnelu<<<gridFor((long long)NN * 128, BS), BS, 0, stream>>>(agg, b1, g1, be1, h, 1, h, 128);
  }
  // ---- GAT layer 2 (1 head, 32 dims, no BN/ELU; bias folded into head) ----
  {
    int tiles = (NN / 16) * (32 / 16);
    k_gemm_wmma_f32<<<gridFor((long long)tiles * 32, BS), BS, 0, stream>>>(h, W2, hW, 32);
    run_attention(a2s, a2d, 1, 32);
  }
  // ---- pooling + head ----
  k_fill_u32<<<gridFor(BG * 33, BS), BS, 0, stream>>>((unsigned*)sums, 0u, BG * 33); // sums + cnt contiguous
  k_pool<<<gridFor((long long)NN * 32, BS), BS, 0, stream>>>(batch, agg, sums, cnt);
  float* outLog = (float*)d_out;
  float* outEmb = (float*)d_out + BG * 10;
  k_head<<<BG, 256, 0, stream>>>(sums, cnt, b2, c0w, c0b, cg0, cb0, c1w, c1b,
                                 l1w, l1b, l2w, l2b, outLog, outEmb);
}